// WCAT_61804579389570
// MI455X (gfx1250) — compile-verified
//
#include <hip/hip_runtime.h>
#include <hip/hip_bf16.h>

// Elementwise 4-bit fake-quant + XOR bit-flip noise + dequant.
// Memory-bound: 768 MiB moved / call -> ~33 us floor at 23.3 TB/s.
// Strategy: b128 non-temporal loads/stores, 4 elems/thread, wave32 blocks.
// Wave-uniform fast path when q_range is a power of two (the harness value
// 1.0 qualifies): divide becomes one exact multiply, ~10 VALU/element.

typedef __attribute__((ext_vector_type(4))) float v4f;
typedef __attribute__((ext_vector_type(4))) int   v4i;

__device__ __forceinline__ float quant_range(const float* __restrict__ qrp, float n_f) {
    // Faithful grad-scale STE arithmetic: qr = (q - q*gs) + q*gs, gs = 1/sqrt(N).
    // Uniform -> fully scalarized by the compiler (runs once per wave).
    float q  = qrp[0];
    float gs = 1.0f / sqrtf(n_f);
    float t  = q * gs;
    return (q - t) + t;
}

// xi already == x/step (bit-exact); finish the quantize/flip/dequant.
__device__ __forceinline__ float qnoise_finish(float xi, int eps, float step) {
    float xr = __builtin_rintf(xi);          // round half-to-even (RNE)
    xr = fminf(fmaxf(xr, -8.0f), 7.0f);      // clamp to [QN, QP]
    int q = (int)xr;                         // exact: integer-valued float
    int up = (q & 15) ^ eps;                 // two's-complement 4b, XOR flips
    int pert = (up << 28) >> 28;             // sign-extend 4 -> 32 bits
    return (float)pert * step;               // dequantize
}

// True when qr = 2^e with e safe so that 8/qr is an exact normal power of two.
// Then x*(8/qr) is bit-identical to (8x)/qr == x/step (exact binary scaling).
__device__ __forceinline__ bool pow2_fastpath_ok(float qr) {
    unsigned u = __float_as_uint(qr);
    unsigned ex = (u >> 23) & 0xFFu;
    return ((u & 0x807FFFFFu) == 0u)         // positive, mantissa == 0
           && (ex >= 3u) && (ex <= 253u);    // 8/qr stays normal & finite
}

__global__ void __launch_bounds__(256) qnoise_vec4_kernel(
    const float* __restrict__ x,
    const float* __restrict__ qrp,
    const int*   __restrict__ eps,
    float*       __restrict__ out,
    int n4, float n_f)
{
    int i = blockIdx.x * blockDim.x + threadIdx.x;
    if (i >= n4) return;

    float qr   = quant_range(qrp, n_f);
    float step = qr * 0.125f;                // qr / 2^(N_BITS-1), exact

    v4f xv = __builtin_nontemporal_load(((const v4f*)x)   + i);  // global_load_b128 nt
    v4i ev = __builtin_nontemporal_load(((const v4i*)eps) + i);  // global_load_b128 nt

    v4f ov;
    if (pow2_fastpath_ok(qr)) {              // uniform scalar branch (s_cbranch)
        float s8 = 8.0f / qr;                // exact: 2^(3-e), once per wave
#pragma unroll
        for (int k = 0; k < 4; ++k)
            ov[k] = qnoise_finish(xv[k] * s8, ev[k], step);
    } else {                                 // general qr: exact IEEE divide
#pragma unroll
        for (int k = 0; k < 4; ++k)
            ov[k] = qnoise_finish((xv[k] * 8.0f) / qr, ev[k], step);
    }

    __builtin_nontemporal_store(ov, ((v4f*)out) + i);            // global_store_b128 nt
}

// Scalar tail for n not divisible by 4 (not hit for 8192*8192, kept for safety).
__global__ void __launch_bounds__(64) qnoise_tail_kernel(
    const float* __restrict__ x,
    const float* __restrict__ qrp,
    const int*   __restrict__ eps,
    float*       __restrict__ out,
    int start, int n, float n_f)
{
    int i = start + blockIdx.x * blockDim.x + threadIdx.x;
    if (i >= n) return;
    float qr   = quant_range(qrp, n_f);
    float step = qr * 0.125f;
    float xv   = __builtin_nontemporal_load(x + i);
    int   ev   = __builtin_nontemporal_load(eps + i);
    __builtin_nontemporal_store(qnoise_finish((xv * 8.0f) / qr, ev, step), out + i);
}

extern "C" void kernel_launch(void* const* d_in, const int* in_sizes, int n_in,
                              void* d_out, int out_size, void* d_ws, size_t ws_size,
                              hipStream_t stream) {
    const float* x   = (const float*)d_in[0];   // [8192*8192] f32
    const float* qrp = (const float*)d_in[1];   // [1] f32 scalar
    const int*   eps = (const int*)d_in[2];     // [8192*8192] i32
    float*       out = (float*)d_out;

    const int n   = in_sizes[0];
    const int n4  = n >> 2;
    const int rem = n - (n4 << 2);
    const float n_f = (float)n;

    if (n4 > 0) {
        const int block = 256;                  // 8 wave32 waves / block
        const int grid  = (n4 + block - 1) / block;
        qnoise_vec4_kernel<<<grid, block, 0, stream>>>(x, qrp, eps, out, n4, n_f);
    }
    if (rem > 0) {
        qnoise_tail_kernel<<<1, 64, 0, stream>>>(x, qrp, eps, out, n4 << 2, n, n_f);
    }
}